// Depthavgpooling_78683800862733
// MI455X (gfx1250) — compile-verified
//
#include <hip/hip_runtime.h>
#include <cstdint>

// Problem constants (from reference)
constexpr int N  = 8,  C  = 64, H = 512, W = 512;
constexpr int Ho = (H - 3) / 2 + 1;   // 255
constexpr int Wo = (W - 3) / 2 + 1;   // 255

// Tiling
constexpr int OH_T = 4;               // output rows per block
constexpr int OW_T = 64;              // output cols per block
constexpr int IH_T = 2 * OH_T + 1;    // 9 input rows
constexpr int IW_T = 2 * OW_T + 1;    // 129 input cols (max)
constexpr int CB   = 32;              // channels per block (C split in 2 -> 2 WGs/WGP fit in 320KB LDS)
constexpr int CC   = 4;               // channels per wave (8 waves * 4 = 32)

typedef unsigned int v4u __attribute__((ext_vector_type(4)));
typedef int          v8i __attribute__((ext_vector_type(8)));
typedef int          v4i __attribute__((ext_vector_type(4)));

// Issue one TDM load of a 3-D tile (tw x th x tz elements, f32) from global to LDS.
// D# bitfield layout per CDNA5 ISA ch.8 (groups 0..3). Tile==tensor dims: no OOB.
// amdgpu-toolchain (clang-23) builtin form: 6 args (v4u, v8i, v4i, v4i, v8i, i32 cpol).
__device__ __forceinline__ void tdm_load_3d_f32(uint32_t lds_off, const void* gptr,
                                                uint32_t tw, uint32_t th, uint32_t tz,
                                                uint64_t stride0, uint64_t stride1) {
  const uint64_t ga = (uint64_t)gptr;
  v4u g0;
  g0[0] = 1u;                                            // count=1, user descriptor
  g0[1] = lds_off;                                       // lds_addr (bytes)
  g0[2] = (uint32_t)(ga & 0xFFFFFFFFu);                  // global_addr[31:0]
  g0[3] = (uint32_t)((ga >> 32) & 0x1FFFFFFu)            // global_addr[56:32]
        | (2u << 30);                                    // type = 2 ("image")
  v8i g1;
  g1[0] = (int)(2u << 16);                               // data_size = 2 -> 4 bytes; mask=0
  g1[1] = (int)((tw & 0xFFFFu) << 16);                   // tensor_dim0[15:0]
  g1[2] = (int)(((tw >> 16) & 0xFFFFu) | ((th & 0xFFFFu) << 16));   // dim0 hi | dim1 lo
  g1[3] = (int)(((th >> 16) & 0xFFFFu) | ((tw & 0xFFFFu) << 16));   // dim1 hi | tile_dim0
  g1[4] = (int)((th & 0xFFFFu) | ((tz & 0xFFFFu) << 16));           // tile_dim1 | tile_dim2
  g1[5] = (int)(uint32_t)(stride0 & 0xFFFFFFFFu);                   // dim0_stride[31:0]
  g1[6] = (int)(((uint32_t)((stride0 >> 32) & 0xFFFFu))             // dim0_stride[47:32]
        | ((uint32_t)(stride1 & 0xFFFFu) << 16));                   // dim1_stride[15:0]
  g1[7] = (int)(uint32_t)((stride1 >> 16) & 0xFFFFFFFFu);           // dim1_stride[47:16]
  v4i g2;
  g2[0] = (int)tz;                                       // tensor_dim2
  g2[1] = 0; g2[2] = 0; g2[3] = 0;                       // dim3 / dim2_stride / tile_dim3 = 0
  v4i g3; g3[0] = 0; g3[1] = 0; g3[2] = 0; g3[3] = 0;
  v8i g4; g4[0] = 0; g4[1] = 0; g4[2] = 0; g4[3] = 0;    // extra descriptor state: unused
  g4[4] = 0; g4[5] = 0; g4[6] = 0; g4[7] = 0;
  __builtin_amdgcn_tensor_load_to_lds(g0, g1, g2, g3, g4, 0);
}

__global__ __launch_bounds__(256)
void depthavgpool_kernel(const float* __restrict__ x,
                         const float* __restrict__ depth,
                         float* __restrict__ out) {
  const int tid = threadIdx.x;
  const int wid = tid >> 5;                       // wave id 0..7
  const int ow0 = blockIdx.x * OW_T;
  const int oh0 = blockIdx.y * OH_T;
  const int n   = blockIdx.z >> 1;
  const int c0b = (blockIdx.z & 1) * CB;          // channel half: 0 or 32
  const int w0  = ow0 * 2;
  const int h0  = oh0 * 2;
  const int tw  = (W - w0) < IW_T ? (W - w0) : IW_T;   // 129 (128 on last col tile)
  const int th  = (H - h0) < IH_T ? (H - h0) : IH_T;   // 9   (8 on last row tile)

  extern __shared__ float smem[];
  float* xs  = smem;                              // [CB][th][tw], pitch = tw
  float* dph = smem + CB * IH_T * IW_T;           // [th][tw]

  // ---- 1) per-wave TDM: DMA 4 channels of the input tile into LDS (async) ----
  {
    const int    cw   = c0b + wid * CC;
    const float* gsrc = x + (((size_t)n * C + cw) * H + h0) * (size_t)W + w0;
    const uint32_t lds_off =
        (uint32_t)(uintptr_t)(xs + (size_t)(wid * CC) * th * tw);
    tdm_load_3d_f32(lds_off, gsrc, (uint32_t)tw, (uint32_t)th, (uint32_t)CC,
                    (uint64_t)W, (uint64_t)H * (uint64_t)W);
  }

  // ---- 2) depth tile -> LDS with ordinary coalesced loads (overlaps the DMA) ----
  {
    const float* dsrc = depth + ((size_t)n * H + h0) * (size_t)W + w0;
    for (int r = 0; r < th; ++r)
      for (int c = tid; c < tw; c += 256)
        dph[r * tw + c] = dsrc[(size_t)r * W + c];
  }
  __syncthreads();

  // ---- 3) per-pixel weights in registers (shared across the block's 32 channels) ----
  const int  ly = tid >> 6, lx = tid & 63;
  const int  oh = oh0 + ly, ow = ow0 + lx;
  const bool valid = (oh < Ho) && (ow < Wo);
  const int  ry = valid ? ly * 2 : 0;             // clamp so inactive threads stay in-bounds
  const int  rx = valid ? lx * 2 : 0;

  float wv[9];
  float den = 0.0f;
  const float dc = dph[(ry + 1) * tw + (rx + 1)];
#pragma unroll
  for (int ky = 0; ky < 3; ++ky)
#pragma unroll
    for (int kx = 0; kx < 3; ++kx) {
      const float d = dph[(ry + ky) * tw + (rx + kx)];
      const float e = __expf(-fabsf(dc - d));
      wv[ky * 3 + kx] = e;
      den += e;
    }
  const float inv = __builtin_amdgcn_rcpf(den);

  // ---- 4) wait for own TDM chunk, then barrier so every chunk is visible ----
  __builtin_amdgcn_s_wait_tensorcnt(0);
  __syncthreads();

  // ---- 5) 32 channels: 9-tap weighted sum from LDS, NT store to global ----
  const size_t cstride  = (size_t)Ho * Wo;
  float* const obase    = out + (((size_t)n * C + c0b) * Ho + oh) * (size_t)Wo + ow;
  const int    rowoff   = ry * tw + rx;
  const int    chstride = th * tw;
#pragma unroll 4
  for (int ch = 0; ch < CB; ++ch) {
    const float* p = xs + ch * chstride + rowoff;
    float num = wv[0] * p[0] + wv[1] * p[1] + wv[2] * p[2];
    p += tw;
    num += wv[3] * p[0] + wv[4] * p[1] + wv[5] * p[2];
    p += tw;
    num += wv[6] * p[0] + wv[7] * p[1] + wv[8] * p[2];
    if (valid)
      __builtin_nontemporal_store(num * inv, obase + (size_t)ch * cstride);
  }
}

extern "C" void kernel_launch(void* const* d_in, const int* in_sizes, int n_in,
                              void* d_out, int out_size, void* d_ws, size_t ws_size,
                              hipStream_t stream) {
  const float* x     = (const float*)d_in[0];
  const float* depth = (const float*)d_in[1];
  float*       out   = (float*)d_out;

  const int colTiles = (Wo + OW_T - 1) / OW_T;    // 4
  const int rowTiles = (Ho + OH_T - 1) / OH_T;    // 64
  dim3 grid(colTiles, rowTiles, N * 2);           // z = batch * channel-half
  dim3 block(256);
  const size_t shmem = (size_t)(CB * IH_T * IW_T + IH_T * IW_T) * sizeof(float); // ~153 KB

  depthavgpool_kernel<<<grid, block, shmem, stream>>>(x, depth, out);
}